// RenderLayer_37357625541288
// MI455X (gfx1250) — compile-verified
//
#include <hip/hip_runtime.h>
#include <stdint.h>

#define H_IMG   224
#define W_IMG   224
#define EPSF    1e-8f
#define N_VERTS 6890
#define N_FACES 13776
#define BATCH   2
#define CH      128                 // faces per LDS chunk (128 * 64B = 8KB)
#define FPAD    13824               // N_FACES rounded up to CH multiple (108 chunks)

typedef __attribute__((ext_vector_type(2))) float v2f;
typedef __attribute__((ext_vector_type(8))) float v8f;

// ---------------------------------------------------------------------------
// Kernel 1: Rodrigues + projection. uv[b][v][2], zz[b][v]
// ---------------------------------------------------------------------------
__global__ void k_project(const float* __restrict__ v,
                          const float* __restrict__ rt,
                          const float* __restrict__ t,
                          const float* __restrict__ cf,
                          const float* __restrict__ cc,
                          float* __restrict__ uv,
                          float* __restrict__ zz) {
    int i = blockIdx.x * blockDim.x + threadIdx.x;
    if (i >= BATCH * N_VERTS) return;
    float r0 = rt[0], r1 = rt[1], r2 = rt[2];
    float th = sqrtf(r0 * r0 + r1 * r1 + r2 * r2 + 1e-12f);
    float kx = r0 / th, ky = r1 / th, kz = r2 / th;
    float c = cosf(th), s = sinf(th), omc = 1.0f - c;
    float R00 = c + omc * kx * kx,      R01 = omc * kx * ky - s * kz, R02 = omc * kx * kz + s * ky;
    float R10 = omc * ky * kx + s * kz, R11 = c + omc * ky * ky,      R12 = omc * ky * kz - s * kx;
    float R20 = omc * kz * kx - s * ky, R21 = omc * kz * ky + s * kx, R22 = c + omc * kz * kz;
    float x = v[i * 3 + 0], y = v[i * 3 + 1], z = v[i * 3 + 2];
    float X = R00 * x + R01 * y + R02 * z + t[0];
    float Y = R10 * x + R11 * y + R12 * z + t[1];
    float Z = R20 * x + R21 * y + R22 * z + t[2];
    float zs = fmaxf(Z, EPSF);
    uv[i * 2 + 0] = X / zs * cf[0] + cc[0];
    uv[i * 2 + 1] = Y / zs * cf[1] + cc[1];
    zz[i] = Z;
}

// ---------------------------------------------------------------------------
// Kernel 2: per-face affine coefficients, padded to FPAD faces per batch:
//   [Ab0 Bb0 Cb0 0 | Ab1 Bb1 Cb1 0 | Ab2 Bb2 Cb2 0 | Az Bz Cz 0]
// b_i(p) = A*px + B*py + C (inv_area folded in); inv_z(p) likewise.
// Invalid/pad faces poisoned so b0 < 0 everywhere -> never inside.
// ---------------------------------------------------------------------------
__global__ void k_facesetup(const int* __restrict__ f,
                            const float* __restrict__ uv,
                            const float* __restrict__ zz,
                            float* __restrict__ coef) {
    int i = blockIdx.x * blockDim.x + threadIdx.x;
    if (i >= BATCH * FPAD) return;
    int b  = i / FPAD;
    int fi = i - b * FPAD;
    float4* o4 = (float4*)(coef + (size_t)i * 16);
    bool valid = (fi < N_FACES);
    float area = 0.f, x0=0,y0=0,x1=0,y1=0,x2=0,y2=0,z0=0,z1=0,z2=0;
    if (valid) {
        int i0 = f[fi * 3 + 0], i1 = f[fi * 3 + 1], i2 = f[fi * 3 + 2];
        const float* uvb = uv + (size_t)b * N_VERTS * 2;
        const float* zb  = zz + (size_t)b * N_VERTS;
        x0 = uvb[i0 * 2]; y0 = uvb[i0 * 2 + 1];
        x1 = uvb[i1 * 2]; y1 = uvb[i1 * 2 + 1];
        x2 = uvb[i2 * 2]; y2 = uvb[i2 * 2 + 1];
        z0 = zb[i0]; z1 = zb[i1]; z2 = zb[i2];
        area = (x1 - x0) * (y2 - y0) - (y1 - y0) * (x2 - x0);
        valid = (fabsf(area) > EPSF) && (z0 > EPSF) && (z1 > EPSF) && (z2 > EPSF);
    }
    if (!valid) {
        o4[0] = make_float4(0.f, 0.f, -1e30f, 0.f);
        o4[1] = make_float4(0.f, 0.f, 0.f, 0.f);
        o4[2] = make_float4(0.f, 0.f, 0.f, 0.f);
        o4[3] = make_float4(0.f, 0.f, 0.f, 0.f);
        return;
    }
    float ia = 1.0f / area;
    float A0 = (y1 - y2) * ia, B0 = (x2 - x1) * ia, C0 = (x1 * y2 - y1 * x2) * ia;
    float A1 = (y2 - y0) * ia, B1 = (x0 - x2) * ia, C1 = (x2 * y0 - y2 * x0) * ia;
    float A2 = (y0 - y1) * ia, B2 = (x1 - x0) * ia, C2 = (x0 * y1 - y0 * x1) * ia;
    float s0 = 1.0f / fmaxf(z0, EPSF);
    float s1 = 1.0f / fmaxf(z1, EPSF);
    float s2 = 1.0f / fmaxf(z2, EPSF);
    o4[0] = make_float4(A0, B0, C0, 0.f);
    o4[1] = make_float4(A1, B1, C1, 0.f);
    o4[2] = make_float4(A2, B2, C2, 0.f);
    o4[3] = make_float4(A0 * s0 + A1 * s1 + A2 * s2,
                        B0 * s0 + B1 * s1 + B2 * s2,
                        C0 * s0 + C1 * s1 + C2 * s2, 0.f);
}

// ---------------------------------------------------------------------------
// CDNA5 async global -> LDS (ASYNCcnt), 16B per lane; in-order completion.
// ---------------------------------------------------------------------------
__device__ __forceinline__ void async_b128_to_lds(uint32_t lds_byte_addr, const void* g) {
    asm volatile("global_load_async_to_lds_b128 %0, %1, off"
                 :: "v"(lds_byte_addr), "v"(g) : "memory");
}
__device__ __forceinline__ void wait_asynccnt_le2() {
    asm volatile("s_wait_asynccnt 2" ::: "memory");
}
__device__ __forceinline__ void wait_asynccnt_0() {
    asm volatile("s_wait_asynccnt 0" ::: "memory");
}

// ---------------------------------------------------------------------------
// Kernel 3: tiled rasterizer, double-buffered async face stream.
// WG = 256 thr = 8 waves; tile = 16x8 pixels; wave w owns tile row w.
// Per chunk per wave: 32 x V_WMMA_F32_16X16X4_F32 (A = 4 faces x
// {b0,b1,b2,inv_z} coeff rows, B = pixel basis (px,py,1,0)); per-lane z-test
// on 2 faces via min3 + single compare; cross-half merge with shfl_xor(16).
// ---------------------------------------------------------------------------
__global__ __launch_bounds__(256) void k_raster(const float* __restrict__ coef,
                                                const float* __restrict__ zz,
                                                const int* __restrict__ f,
                                                const float* __restrict__ vc,
                                                const float* __restrict__ bg,
                                                float* __restrict__ out) {
    __shared__ float lds[2 * CH * 16];          // two 8KB buffers
    const int b    = blockIdx.z;
    const int tid  = threadIdx.x;
    const int wave = tid >> 5;
    const int lane = tid & 31;
    const int N    = lane & 15;
    const int hi   = lane >> 4;                 // 0 = lanes 0-15, 1 = lanes 16-31
    const int px_i = blockIdx.x * 16 + N;
    const int py_i = blockIdx.y * 8 + wave;
    const float pxf = (float)px_i + 0.5f;
    const float pyf = (float)py_i + 0.5f;

    // B matrix 4x16: lane<16 column N holds K0=px, K1=py; hi lanes K2=1, K3=0
    v2f bmat;
    bmat.x = hi ? 1.0f : pxf;
    bmat.y = hi ? 0.0f : pyf;

    float best = EPSF;                          // max inv_z (== min zp); '>' keeps lowest face on ties
    int   bf   = -1;

    const float* ct = coef + (size_t)b * FPAD * 16;
    const uint32_t lds_base = (uint32_t)(uintptr_t)&lds[0];
    const int nchunk = FPAD / CH;               // 108, no tail (table padded+poisoned)
    const int M = lane & 15;                    // A-matrix row owned by this lane

    // prologue: chunk 0 -> buffer 0
    for (int s = tid; s < CH * 4; s += 256) {
        async_b128_to_lds(lds_base + (uint32_t)s * 16u,
                          (const void*)(ct + (size_t)(s >> 2) * 16 + (size_t)(s & 3) * 4));
    }

    for (int c = 0; c < nchunk; ++c) {
        const bool more = (c + 1 < nchunk);
        if (more) {
            // stream next chunk into the other buffer while this one computes
            const uint32_t boff = (uint32_t)(((c + 1) & 1) * CH * 64);
            const float* src0 = ct + (size_t)(c + 1) * CH * 16;
            if (c + 2 < nchunk)
                __builtin_prefetch(ct + (size_t)(c + 2) * CH * 16, 0, 1);
            for (int s = tid; s < CH * 4; s += 256) {
                async_b128_to_lds(lds_base + boff + (uint32_t)s * 16u,
                                  (const void*)(src0 + (size_t)(s >> 2) * 16 + (size_t)(s & 3) * 4));
            }
            wait_asynccnt_le2();                // oldest 2 (= this chunk) done, in order
        } else {
            wait_asynccnt_0();
        }
        __syncthreads();

        const float* buf = lds + (c & 1) * CH * 16;
        const int fb = c * CH;
#pragma unroll 4
        for (int g = 0; g < CH / 4; ++g) {      // 4 faces per WMMA
            // A 16x4: lane<16 holds K0,K1 (A,B coeffs); hi lanes K2,K3 (C,0)
            const float2 av = *(const float2*)(buf + g * 64 + M * 4 + (hi ? 2 : 0));
            v2f a; a.x = av.x; a.y = av.y;
            v8f cacc = {};
            cacc = __builtin_amdgcn_wmma_f32_16x16x4_f32(
                false, a, false, bmat, (short)0, cacc, false, false);
            const int fgBase = fb + g * 4 + (hi ? 2 : 0);
            {   // face fgBase: rows 0..3 (lo) / 8..11 (hi)
                float m = fminf(fminf(cacc[0], cacc[1]), cacc[2]);   // v_min3
                float cand = (m >= 0.f) ? cacc[3] : -1.0f;
                if (cand > best) { best = cand; bf = fgBase; }
            }
            {   // face fgBase+1: rows 4..7 (lo) / 12..15 (hi)
                float m = fminf(fminf(cacc[4], cacc[5]), cacc[6]);
                float cand = (m >= 0.f) ? cacc[7] : -1.0f;
                if (cand > best) { best = cand; bf = fgBase + 1; }
            }
        }
        __syncthreads();
    }

    // merge the two lane-halves (each saw a disjoint face subset for pixel N)
    float ob = __shfl_xor(best, 16, 32);
    int   of = __shfl_xor(bf,   16, 32);
    if (of >= 0 && (ob > best || (ob == best && (bf < 0 || of < bf)))) { best = ob; bf = of; }

    if (!hi && px_i < W_IMG && py_i < H_IMG) {
        float cr, cg, cb;
        if (bf >= 0) {
            const float* o = ct + (size_t)bf * 16;
            float b0 = o[0] * pxf + o[1] * pyf + o[2];
            float b1 = o[4] * pxf + o[5] * pyf + o[6];
            float b2 = o[8] * pxf + o[9] * pyf + o[10];
            float zmin = 1.0f / best;           // best > EPS guaranteed
            int i0 = f[bf * 3 + 0], i1 = f[bf * 3 + 1], i2 = f[bf * 3 + 2];
            const float* zb = zz + (size_t)b * N_VERTS;
            float p0 = b0 / fmaxf(zb[i0], EPSF);
            float p1 = b1 / fmaxf(zb[i1], EPSF);
            float p2 = b2 / fmaxf(zb[i2], EPSF);
            cr = (p0 * vc[i0 * 3 + 0] + p1 * vc[i1 * 3 + 0] + p2 * vc[i2 * 3 + 0]) * zmin;
            cg = (p0 * vc[i0 * 3 + 1] + p1 * vc[i1 * 3 + 1] + p2 * vc[i2 * 3 + 1]) * zmin;
            cb = (p0 * vc[i0 * 3 + 2] + p1 * vc[i1 * 3 + 2] + p2 * vc[i2 * 3 + 2]) * zmin;
        } else {
            cr = bg[0]; cg = bg[1]; cb = bg[2];
        }
        size_t oo = (((size_t)b * H_IMG + py_i) * W_IMG + px_i) * 3;
        out[oo + 0] = cr;
        out[oo + 1] = cg;
        out[oo + 2] = cb;
    }
}

// ---------------------------------------------------------------------------
extern "C" void kernel_launch(void* const* d_in, const int* in_sizes, int n_in,
                              void* d_out, int out_size, void* d_ws, size_t ws_size,
                              hipStream_t stream) {
    const float* v   = (const float*)d_in[0];
    const float* vc  = (const float*)d_in[1];
    const float* bg  = (const float*)d_in[2];
    const float* cf  = (const float*)d_in[3];
    const float* cc  = (const float*)d_in[4];
    const float* ct  = (const float*)d_in[5];
    const float* crt = (const float*)d_in[6];
    const int*   f   = (const int*)d_in[7];
    float* out = (float*)d_out;

    float* ws   = (float*)d_ws;
    float* uv   = ws;                                  // BATCH*N_VERTS*2
    float* zzv  = uv + (size_t)BATCH * N_VERTS * 2;    // BATCH*N_VERTS
    float* coef = zzv + (size_t)BATCH * N_VERTS;       // BATCH*FPAD*16

    {
        int n = BATCH * N_VERTS;
        k_project<<<(n + 255) / 256, 256, 0, stream>>>(v, crt, ct, cf, cc, uv, zzv);
    }
    {
        int n = BATCH * FPAD;
        k_facesetup<<<(n + 255) / 256, 256, 0, stream>>>(f, uv, zzv, coef);
    }
    {
        dim3 grid((W_IMG + 15) / 16, (H_IMG + 7) / 8, BATCH);
        k_raster<<<grid, 256, 0, stream>>>(coef, zzv, f, vc, bg, out);
    }
}